// NostroMinst_86036784873945
// MI455X (gfx1250) — compile-verified
//
#include <hip/hip_runtime.h>
#include <hip/hip_bf16.h>
#include <math.h>

typedef __attribute__((ext_vector_type(2))) float v2f;
typedef __attribute__((ext_vector_type(4))) float v4f;
typedef __attribute__((ext_vector_type(8))) float v8f;

#define Bn    32
#define HW    4096      // 64x64
#define Wdim  64
#define C1n   32
#define C2n   64
#define KK    5
#define FC1n  512

// workspace layout (float offsets)
#define OFF_M1 0                           // C1n*HW      = 131072
#define OFF_M2 (OFF_M1 + C1n*HW)           // C2n*HW      = 262144
#define OFF_AX (OFF_M2 + C2n*HW)           // Bn*HW       = 131072
#define OFF_Y1 (OFF_AX + Bn*HW)            // Bn*FC1n     = 16384
#define OFF_G  (OFF_Y1 + Bn*FC1n)          // FC1n*HW     = 2097152

// ---------------------------------------------------------------------------
// K1: M1[c1,p] = |FFT2(pad(w1))|  via direct 25-term DFT (kernel is 5x5)
// ---------------------------------------------------------------------------
__global__ void k_dft_mag1(const float* __restrict__ w1r,
                           const float* __restrict__ w1i,
                           float* __restrict__ M1) {
  int tid = blockIdx.x * blockDim.x + threadIdx.x;      // C1n*HW threads
  int c1 = tid >> 12;
  int p  = tid & (HW - 1);
  int h  = p >> 6, w = p & 63;
  const float* wr = w1r + c1 * (KK * KK);
  const float* wi = w1i + c1 * (KK * KK);
  float re = 0.f, im = 0.f;
  const float k2pi = 6.28318530717958647692f / 64.0f;
#pragma unroll
  for (int u = 0; u < KK; ++u)
#pragma unroll
    for (int v = 0; v < KK; ++v) {
      float th = k2pi * (float)(h * u + w * v);
      float s, c;
      __sincosf(th, &s, &c);
      float ar = wr[u * KK + v], ai = wi[u * KK + v];
      // (ar + i*ai) * (c - i*s)
      re += ar * c + ai * s;
      im += ai * c - ar * s;
    }
  M1[c1 * HW + p] = sqrtf(re * re + im * im);
}

// ---------------------------------------------------------------------------
// K2: M2[c2,p] = | sum_c1 M1[c1,p] * DFT(w2[c2,c1]) |
// Reordered: for each (u,v): P = sum_c1 M1*w2[uv]; then one twiddle per (u,v).
// One block per (c2, 256-pixel chunk); w2[c2] slice (32*25*2 fp32) in LDS.
// ---------------------------------------------------------------------------
__global__ void k_dft_mag2(const float* __restrict__ w2r,
                           const float* __restrict__ w2i,
                           const float* __restrict__ M1,
                           float* __restrict__ M2) {
  __shared__ float swr[C1n * KK * KK];   // 800 floats
  __shared__ float swi[C1n * KK * KK];
  int c2    = blockIdx.x >> 4;           // 64 c2 * 16 chunks
  int chunk = blockIdx.x & 15;
  int tid   = threadIdx.x;
  for (int i = tid; i < C1n * KK * KK; i += blockDim.x) {
    swr[i] = w2r[c2 * C1n * KK * KK + i];
    swi[i] = w2i[c2 * C1n * KK * KK + i];
  }
  __syncthreads();

  int p = chunk * 256 + tid;
  int h = p >> 6, w = p & 63;
  float m1v[C1n];
#pragma unroll
  for (int c = 0; c < C1n; ++c) m1v[c] = M1[c * HW + p];

  float re = 0.f, im = 0.f;
  const float k2pi = 6.28318530717958647692f / 64.0f;
#pragma unroll
  for (int u = 0; u < KK; ++u)
#pragma unroll
    for (int v = 0; v < KK; ++v) {
      int uv = u * KK + v;
      float pr = 0.f, pi = 0.f;
#pragma unroll
      for (int c = 0; c < C1n; ++c) {
        pr = fmaf(m1v[c], swr[c * KK * KK + uv], pr);
        pi = fmaf(m1v[c], swi[c * KK * KK + uv], pi);
      }
      float th = k2pi * (float)(h * u + w * v);
      float s, c;
      __sincosf(th, &s, &c);
      re += pr * c + pi * s;
      im += pi * c - pr * s;
    }
  M2[c2 * HW + p] = sqrtf(re * re + im * im);
}

// ---------------------------------------------------------------------------
// K3: AX[b,p] = |x[b,p]|
// ---------------------------------------------------------------------------
__global__ void k_absx(const float* __restrict__ x, float* __restrict__ ax) {
  int i = blockIdx.x * blockDim.x + threadIdx.x;   // Bn*HW
  ax[i] = fabsf(x[i]);
}

// ---------------------------------------------------------------------------
// K4: the 536 MB stream.  G[f,p] = sum_c2 M2[c2,p] * fc1_w[f, c2, p]
// Block = (f, 1024-pixel chunk); thread handles 4 pixels (float4).
// fc1_w can never fit L2 (536MB > 192MB) -> nontemporal loads.
// ---------------------------------------------------------------------------
__global__ void k_gbuild(const float* __restrict__ fc1_w,
                         const float* __restrict__ M2,
                         float* __restrict__ G) {
  int f     = blockIdx.x >> 2;            // 512 f
  int chunk = blockIdx.x & 3;             // 4 chunks of 1024 px
  int p0    = chunk * 1024 + threadIdx.x * 4;
  const float* wbase = fc1_w + (size_t)f * (C2n * HW) + p0;
  v4f acc = {0.f, 0.f, 0.f, 0.f};
#pragma unroll 4
  for (int c2 = 0; c2 < C2n; ++c2) {
    const v4f* wp = (const v4f*)(wbase + (size_t)c2 * HW);
    __builtin_prefetch(wbase + (size_t)(c2 + 1) * HW, 0, 0);  // global_prefetch
    v4f wv = __builtin_nontemporal_load(wp);                  // th:NT stream
    v4f mv = *(const v4f*)(M2 + c2 * HW + p0);
    acc += wv * mv;
  }
  *(v4f*)(G + (size_t)f * HW + p0) = acc;
}

// ---------------------------------------------------------------------------
// K5: Y1(32x512) = AX(32x4096) @ G^T(4096x512) + fc1_b   via WMMA f32 16x16x4
// One wave per 16x16 output tile; 64 tiles; K-loop 4096 in steps of 4.
// A lane layout (ISA 7.12.2, 32-bit A 16x4): m = lane%16, K = {0,1}|{2,3}.
// ---------------------------------------------------------------------------
__global__ void k_fc1_wmma(const float* __restrict__ AX,
                           const float* __restrict__ G,
                           const float* __restrict__ fc1_b,
                           float* __restrict__ Y1) {
  int tile = blockIdx.x;            // 64 tiles = 2 (batch) x 32 (features)
  int tm = tile & 1;                // batch tile
  int tn = tile >> 1;               // feature tile
  int lane = threadIdx.x;
  int l15  = lane & 15;
  int koff = (lane >> 4) << 1;      // 0 for lanes 0-15, 2 for lanes 16-31

  const float* arow = AX + (size_t)(tm * 16 + l15) * HW + koff;
  const float* brow = G  + (size_t)(tn * 16 + l15) * HW + koff;

  v8f acc = {0.f, 0.f, 0.f, 0.f, 0.f, 0.f, 0.f, 0.f};
#pragma unroll 4
  for (int kk = 0; kk < HW; kk += 4) {
    v2f a = *(const v2f*)(arow + kk);
    v2f b = *(const v2f*)(brow + kk);
    acc = __builtin_amdgcn_wmma_f32_16x16x4_f32(
        /*neg_a=*/false, a, /*neg_b=*/false, b,
        /*c_mod=*/(short)0, acc, /*reuse_a=*/false, /*reuse_b=*/false);
  }

  int fcol = tn * 16 + l15;
  float bias = fc1_b[fcol];
  int mbase = tm * 16 + ((lane >> 4) << 3);   // +8 for lanes 16-31
#pragma unroll
  for (int r = 0; r < 8; ++r)
    Y1[(size_t)(mbase + r) * FC1n + fcol] = acc[r] + bias;
}

// ---------------------------------------------------------------------------
// K6: out(32x10) = Y1(32x512) @ fc2_w^T + fc2_b
// ---------------------------------------------------------------------------
__global__ void k_fc2(const float* __restrict__ Y1,
                      const float* __restrict__ fc2_w,
                      const float* __restrict__ fc2_b,
                      float* __restrict__ out) {
  int t = blockIdx.x * blockDim.x + threadIdx.x;   // 320 threads
  int b = t / 10, j = t % 10;
  const float* y = Y1 + (size_t)b * FC1n;
  const float* w = fc2_w + (size_t)j * FC1n;
  float acc = fc2_b[j];
  for (int f = 0; f < FC1n; ++f) acc = fmaf(y[f], w[f], acc);
  out[b * 10 + j] = acc;
}

// ---------------------------------------------------------------------------
extern "C" void kernel_launch(void* const* d_in, const int* in_sizes, int n_in,
                              void* d_out, int out_size, void* d_ws, size_t ws_size,
                              hipStream_t stream) {
  const float* x     = (const float*)d_in[0];
  const float* w1r   = (const float*)d_in[1];
  const float* w1i   = (const float*)d_in[2];
  const float* w2r   = (const float*)d_in[3];
  const float* w2i   = (const float*)d_in[4];
  const float* fc1_w = (const float*)d_in[5];
  const float* fc1_b = (const float*)d_in[6];
  const float* fc2_w = (const float*)d_in[7];
  const float* fc2_b = (const float*)d_in[8];
  float* out = (float*)d_out;

  float* ws = (float*)d_ws;
  float* M1 = ws + OFF_M1;
  float* M2 = ws + OFF_M2;
  float* AX = ws + OFF_AX;
  float* Y1 = ws + OFF_Y1;
  float* G  = ws + OFF_G;

  k_dft_mag1<<<(C1n * HW) / 256, 256, 0, stream>>>(w1r, w1i, M1);
  k_dft_mag2<<<C2n * 16, 256, 0, stream>>>(w2r, w2i, M1, M2);
  k_absx<<<(Bn * HW) / 256, 256, 0, stream>>>(x, AX);
  k_gbuild<<<FC1n * 4, 256, 0, stream>>>(fc1_w, M2, G);
  k_fc1_wmma<<<64, 32, 0, stream>>>(AX, G, fc1_b, Y1);
  k_fc2<<<10, 32, 0, stream>>>(Y1, fc2_w, fc2_b, out);
}